// GAT_76725295775759
// MI455X (gfx1250) — compile-verified
//
#include <hip/hip_runtime.h>
#include <hip/hip_bf16.h>
#include <math.h>

#define NNODES 50000
#define NEDGES 800000
#define NHEADS 4
#define EPSBN 1e-5f

typedef __attribute__((ext_vector_type(2))) float v2f;
typedef __attribute__((ext_vector_type(8))) float v8f;

__device__ __forceinline__ unsigned enc_f32(float f) {
    unsigned u = __float_as_uint(f);
    return (u & 0x80000000u) ? ~u : (u | 0x80000000u);   // monotone float->uint
}
__device__ __forceinline__ float dec_f32(unsigned u) {
    return (u & 0x80000000u) ? __uint_as_float(u & 0x7FFFFFFFu) : __uint_as_float(~u);
}
__device__ __forceinline__ float lrelu(float x) { return x > 0.f ? x : 0.2f * x; }

// ---------------------------------------------------------------------------
// fp32 GEMM via V_WMMA_F32_16X16X4_F32.  C[M,Nc] = A[M,K] @ B[K,Nc].
// grid.x = M/16, (grid.y * blockDim.y) = Nc/16, block = (32, ny).
// A 16x4 fragment: lanes 0-15 rows M, K={0,1}; lanes 16-31 rows M, K={2,3}.
// B 4x16 fragment: VGPR0 lanes0-15=K0 / lanes16-31=K2; VGPR1 = K1/K3.
// C: vgpr v, lane l -> row = v + (l>=16 ? 8:0), col = l&15.
// ---------------------------------------------------------------------------
__global__ void wmma_gemm_f32(const float* __restrict__ A, const float* __restrict__ B,
                              float* __restrict__ C, int K, int Nc) {
    const int lane = threadIdx.x;                      // 0..31, one wave per tile
    const int tm   = blockIdx.x;
    const int tn   = blockIdx.y * blockDim.y + threadIdx.y;
    const int row  = tm * 16 + (lane & 15);
    const int col  = tn * 16 + (lane & 15);
    const int koff = (lane >> 4) << 1;                 // 0 or 2
    const float* ap = A + (size_t)row * K + koff;
    const float* bp = B + (size_t)koff * Nc + col;
    v8f acc = {};
    for (int kk = 0; kk < K; kk += 4) {
        v2f a, b;
        a.x = ap[kk];
        a.y = ap[kk + 1];
        b.x = bp[(size_t)kk * Nc];
        b.y = bp[(size_t)(kk + 1) * Nc];
        acc = __builtin_amdgcn_wmma_f32_16x16x4_f32(false, a, false, b,
                                                    (short)0, acc, false, false);
    }
    const int crow = tm * 16 + (lane >> 4) * 8;
#pragma unroll
    for (int v = 0; v < 8; ++v)
        C[(size_t)(crow + v) * Nc + col] = acc[v];
}

// ---------------------------------------------------------------------------
// Per-(node,head): attention logits a_src.h, a_dst.h and self-loop max seed.
// ---------------------------------------------------------------------------
__global__ void alpha_prep(const float* __restrict__ h, int C,
                           const float* __restrict__ a_src, const float* __restrict__ a_dst,
                           float* __restrict__ als, float* __restrict__ ald,
                           unsigned* __restrict__ menc) {
    int t = blockIdx.x * blockDim.x + threadIdx.x;
    if (t >= NNODES * NHEADS) return;
    int i = t >> 2, hh = t & 3;
    const float* hp = h + (size_t)i * (NHEADS * C) + (size_t)hh * C;
    const float* as = a_src + hh * C;
    const float* ad = a_dst + hh * C;
    float s = 0.f, d = 0.f;
    for (int c = 0; c < C; ++c) { s = fmaf(hp[c], as[c], s); d = fmaf(hp[c], ad[c], d); }
    als[t] = s;
    ald[t] = d;
    menc[t] = enc_f32(lrelu(s + d));                   // self-loop seeds the segment max
}

__global__ void edge_max(const int* __restrict__ ei, const float* __restrict__ als,
                         const float* __restrict__ ald, unsigned* __restrict__ menc) {
    int e = blockIdx.x * blockDim.x + threadIdx.x;
    if (e >= NEDGES) return;
    int src = ei[e], dst = ei[NEDGES + e];
    float4 s4 = ((const float4*)als)[src];
    float4 d4 = ((const float4*)ald)[dst];
    unsigned* mp = menc + (size_t)dst * 4;
    atomicMax(mp + 0, enc_f32(lrelu(s4.x + d4.x)));
    atomicMax(mp + 1, enc_f32(lrelu(s4.y + d4.y)));
    atomicMax(mp + 2, enc_f32(lrelu(s4.z + d4.z)));
    atomicMax(mp + 3, enc_f32(lrelu(s4.w + d4.w)));
}

// exp of self-loop logit; seeds the softmax denominator.
__global__ void node_softmax_init(const float* __restrict__ als, const float* __restrict__ ald,
                                  const unsigned* __restrict__ menc,
                                  float* __restrict__ exs, float* __restrict__ ssum) {
    int t = blockIdx.x * blockDim.x + threadIdx.x;
    if (t >= NNODES * NHEADS) return;
    float m  = dec_f32(menc[t]);
    float ex = expf(lrelu(als[t] + ald[t]) - m);
    exs[t]  = ex;
    ssum[t] = ex;
}

__global__ void edge_expsum(const int* __restrict__ ei, const float* __restrict__ als,
                            const float* __restrict__ ald, const unsigned* __restrict__ menc,
                            float* __restrict__ exed, float* __restrict__ ssum) {
    int e = blockIdx.x * blockDim.x + threadIdx.x;
    if (e >= NEDGES) return;
    int src = ei[e], dst = ei[NEDGES + e];
    float4 s4 = ((const float4*)als)[src];
    float4 d4 = ((const float4*)ald)[dst];
    uint4  m4 = ((const uint4*)menc)[dst];
    float ex0 = expf(lrelu(s4.x + d4.x) - dec_f32(m4.x));
    float ex1 = expf(lrelu(s4.y + d4.y) - dec_f32(m4.y));
    float ex2 = expf(lrelu(s4.z + d4.z) - dec_f32(m4.z));
    float ex3 = expf(lrelu(s4.w + d4.w) - dec_f32(m4.w));
    float4 r = make_float4(ex0, ex1, ex2, ex3);
    ((float4*)exed)[e] = r;
    float* sp = ssum + (size_t)dst * 4;
    atomicAdd(sp + 0, ex0);
    atomicAdd(sp + 1, ex1);
    atomicAdd(sp + 2, ex2);
    atomicAdd(sp + 3, ex3);
}

// agg init = self-loop message (concat layout: col = head*16 + c)
__global__ void agg_init_cat(const float* __restrict__ h, const float* __restrict__ exs,
                             const float* __restrict__ ssum, float* __restrict__ agg) {
    int t = blockIdx.x * blockDim.x + threadIdx.x;
    if (t >= NNODES * 64) return;
    int i = t >> 6, c = t & 63, hh = c >> 4;
    float alpha = exs[(size_t)i * 4 + hh] / ssum[(size_t)i * 4 + hh];
    agg[t] = h[t] * alpha;
}

__global__ void edge_msg_cat(const int* __restrict__ ei, const float* __restrict__ h,
                             const float* __restrict__ exed, const float* __restrict__ ssum,
                             float* __restrict__ agg) {
    long long t = (long long)blockIdx.x * blockDim.x + threadIdx.x;
    if (t >= (long long)NEDGES * 64) return;
    int e = (int)(t >> 6), c = (int)(t & 63), hh = c >> 4;
    int src = ei[e], dst = ei[NEDGES + e];
    float alpha = exed[(size_t)e * 4 + hh] / ssum[(size_t)dst * 4 + hh];
    atomicAdd(agg + (size_t)dst * 64 + c, alpha * h[(size_t)src * 64 + c]);
}

// layer 2: concat=False -> mean over heads folded in (x0.25); h is [N, 4*64]
__global__ void agg_init_mean(const float* __restrict__ h, const float* __restrict__ exs,
                              const float* __restrict__ ssum, float* __restrict__ agg) {
    int t = blockIdx.x * blockDim.x + threadIdx.x;
    if (t >= NNODES * 64) return;
    int i = t >> 6, c = t & 63;
    const float* hp = h + (size_t)i * 256;
    float v = 0.f;
#pragma unroll
    for (int hh = 0; hh < 4; ++hh)
        v = fmaf(exs[(size_t)i * 4 + hh] / ssum[(size_t)i * 4 + hh], hp[hh * 64 + c], v);
    agg[t] = 0.25f * v;
}

__global__ void edge_msg_mean(const int* __restrict__ ei, const float* __restrict__ h,
                              const float* __restrict__ exed, const float* __restrict__ ssum,
                              float* __restrict__ agg) {
    long long t = (long long)blockIdx.x * blockDim.x + threadIdx.x;
    if (t >= (long long)NEDGES * 64) return;
    int e = (int)(t >> 6), c = (int)(t & 63);
    int src = ei[e], dst = ei[NEDGES + e];
    const float* hp = h + (size_t)src * 256;
    float v = 0.f;
#pragma unroll
    for (int hh = 0; hh < 4; ++hh)
        v = fmaf(exed[(size_t)e * 4 + hh] / ssum[(size_t)dst * 4 + hh], hp[hh * 64 + c], v);
    atomicAdd(agg + (size_t)dst * 64 + c, 0.25f * v);
}

__global__ void finalize_bn_relu(const float* __restrict__ agg, const float* __restrict__ convb,
                                 const float* __restrict__ skip, const float* __restrict__ skipb,
                                 const float* __restrict__ g, const float* __restrict__ b,
                                 const float* __restrict__ m, const float* __restrict__ v,
                                 float* __restrict__ xout) {
    int t = blockIdx.x * blockDim.x + threadIdx.x;
    if (t >= NNODES * 64) return;
    int c = t & 63;
    float val = agg[t] + convb[c] + skip[t] + skipb[c];
    val = (val - m[c]) * rsqrtf(v[c] + EPSBN) * g[c] + b[c];
    xout[t] = fmaxf(val, 0.f);
}

__global__ void finalize_out(const float* __restrict__ agg, const float* __restrict__ convb,
                             const float* __restrict__ skip, const float* __restrict__ skipb,
                             float* __restrict__ out) {
    int t = blockIdx.x * blockDim.x + threadIdx.x;
    if (t >= NNODES * 64) return;
    int c = t & 63;
    out[t] = agg[t] + convb[c] + skip[t] + skipb[c];
}

extern "C" void kernel_launch(void* const* d_in, const int* in_sizes, int n_in,
                              void* d_out, int out_size, void* d_ws, size_t ws_size,
                              hipStream_t stream) {
    const float* x  = (const float*)d_in[0];
    const int*   ei = (const int*)d_in[1];
    const float *c0W = (const float*)d_in[2],  *c0as = (const float*)d_in[3],
                *c0ad = (const float*)d_in[4], *c0b = (const float*)d_in[5],
                *s0W = (const float*)d_in[6],  *s0b = (const float*)d_in[7],
                *b0g = (const float*)d_in[8],  *b0b = (const float*)d_in[9],
                *b0m = (const float*)d_in[10], *b0v = (const float*)d_in[11];
    const float *c1W = (const float*)d_in[12], *c1as = (const float*)d_in[13],
                *c1ad = (const float*)d_in[14],*c1b = (const float*)d_in[15],
                *s1W = (const float*)d_in[16], *s1b = (const float*)d_in[17],
                *b1g = (const float*)d_in[18], *b1b = (const float*)d_in[19],
                *b1m = (const float*)d_in[20], *b1v = (const float*)d_in[21];
    const float *c2W = (const float*)d_in[22], *c2as = (const float*)d_in[23],
                *c2ad = (const float*)d_in[24],*c2b = (const float*)d_in[25],
                *s2W = (const float*)d_in[26], *s2b = (const float*)d_in[27];
    float* out = (float*)d_out;

    // workspace layout (floats)
    float* ws = (float*)d_ws;
    size_t o = 0;
    float*    f_h   = ws + o; o += (size_t)NNODES * 256;   // conv output (max width, layer 2)
    float*    f_x   = ws + o; o += (size_t)NNODES * 64;    // current node features
    float*    f_sk  = ws + o; o += (size_t)NNODES * 64;    // skip projection
    float*    f_agg = ws + o; o += (size_t)NNODES * 64;    // aggregated messages
    float*    f_als = ws + o; o += (size_t)NNODES * 4;
    float*    f_ald = ws + o; o += (size_t)NNODES * 4;
    unsigned* f_m   = (unsigned*)(ws + o); o += (size_t)NNODES * 4;
    float*    f_s   = ws + o; o += (size_t)NNODES * 4;
    float*    f_exs = ws + o; o += (size_t)NNODES * 4;
    float*    f_ex  = ws + o; o += (size_t)NEDGES * 4;

    const dim3 gblk(32, 4, 1);                 // 4 waves, each owns one 16-col tile
    const int  TB = 256;
    const int  nNH  = (NNODES * NHEADS + TB - 1) / TB;
    const int  nN64 = (NNODES * 64) / TB;      // exact
    const int  nE   = (NEDGES + TB - 1) / TB;
    const int  nE64 = (int)(((long long)NEDGES * 64) / TB);

    // ---------------- layers 0 and 1 (concat=True, width 64) ----------------
    const float *inX = x;  int K = 128;
    const float *cW[2] = {c0W, c1W}, *cas[2] = {c0as, c1as}, *cad[2] = {c0ad, c1ad},
                *cb[2] = {c0b, c1b}, *sW[2] = {s0W, s1W}, *sb[2] = {s0b, s1b},
                *bg[2] = {b0g, b1g}, *bb[2] = {b0b, b1b}, *bm[2] = {b0m, b1m},
                *bv[2] = {b0v, b1v};
    for (int L = 0; L < 2; ++L) {
        wmma_gemm_f32<<<dim3(NNODES / 16, 1), gblk, 0, stream>>>(inX, cW[L], f_h, K, 64);
        wmma_gemm_f32<<<dim3(NNODES / 16, 1), gblk, 0, stream>>>(inX, sW[L], f_sk, K, 64);
        alpha_prep<<<nNH, TB, 0, stream>>>(f_h, 16, cas[L], cad[L], f_als, f_ald, f_m);
        edge_max<<<nE, TB, 0, stream>>>(ei, f_als, f_ald, f_m);
        node_softmax_init<<<nNH, TB, 0, stream>>>(f_als, f_ald, f_m, f_exs, f_s);
        edge_expsum<<<nE, TB, 0, stream>>>(ei, f_als, f_ald, f_m, f_ex, f_s);
        agg_init_cat<<<nN64, TB, 0, stream>>>(f_h, f_exs, f_s, f_agg);
        edge_msg_cat<<<nE64, TB, 0, stream>>>(ei, f_h, f_ex, f_s, f_agg);
        finalize_bn_relu<<<nN64, TB, 0, stream>>>(f_agg, cb[L], f_sk, sb[L],
                                                  bg[L], bb[L], bm[L], bv[L], f_x);
        inX = f_x;  K = 64;
    }

    // ---------------- layer 2 (concat=False, mean over heads) ----------------
    wmma_gemm_f32<<<dim3(NNODES / 16, 4), gblk, 0, stream>>>(f_x, c2W, f_h, 64, 256);
    wmma_gemm_f32<<<dim3(NNODES / 16, 1), gblk, 0, stream>>>(f_x, s2W, f_sk, 64, 64);
    alpha_prep<<<nNH, TB, 0, stream>>>(f_h, 64, c2as, c2ad, f_als, f_ald, f_m);
    edge_max<<<nE, TB, 0, stream>>>(ei, f_als, f_ald, f_m);
    node_softmax_init<<<nNH, TB, 0, stream>>>(f_als, f_ald, f_m, f_exs, f_s);
    edge_expsum<<<nE, TB, 0, stream>>>(ei, f_als, f_ald, f_m, f_ex, f_s);
    agg_init_mean<<<nN64, TB, 0, stream>>>(f_h, f_exs, f_s, f_agg);
    edge_msg_mean<<<nE64, TB, 0, stream>>>(ei, f_h, f_ex, f_s, f_agg);
    finalize_out<<<nN64, TB, 0, stream>>>(f_agg, c2b, f_sk, s2b, out);
}